// GATLayer_58402965291024
// MI455X (gfx1250) — compile-verified
//
#include <hip/hip_runtime.h>
#include <math.h>

#define IN_DIM   128
#define OUT_DIM  64
#define NEG_SLOPE 0.2f

typedef float v2f __attribute__((ext_vector_type(2)));
typedef float v8f __attribute__((ext_vector_type(8)));

__device__ __forceinline__ float leaky(float e) {
  return (e > 0.f) ? e : NEG_SLOPE * e;
}

// ---------------------------------------------------------------------------
// Kernel 0: transpose W (128x64) -> Wt (64x128) so both WMMA operand
// fragments are contiguous 8-byte loads. 32 KB, stays L1/L2 resident.
// ---------------------------------------------------------------------------
__global__ void k_wt(const float* __restrict__ W, float* __restrict__ Wt) {
  int i = blockIdx.x * blockDim.x + threadIdx.x;   // i over IN_DIM*OUT_DIM
  if (i >= IN_DIM * OUT_DIM) return;
  int k = i / OUT_DIM;
  int c = i - k * OUT_DIM;
  Wt[(size_t)c * IN_DIM + k] = W[i];
}

// ---------------------------------------------------------------------------
// Kernel 1: feat = h @ W via V_WMMA_F32_16X16X4_F32 (exact fp32 matmul).
// 256 threads = 8 waves per block; each block computes a 32x64 tile of feat.
// fp32 WMMA operand layout (ISA 7.12.2):
//   A 16x4 : 2 VGPRs/lane. lanes 0-15 -> M=lane, K={0,1}; lanes 16-31 -> K={2,3}
//   B 4x16 : 2 VGPRs/lane. lanes 0-15 -> N=lane, K={0,1}; lanes 16-31 -> K={2,3}
//   C 16x16: 8 VGPRs. VGPR r -> M=r (lanes 0-15), M=r+8 (lanes 16-31), N=lane&15
// Rows are clamped (not predicated) so loads are unconditional; compiler
// software-pipelines ~22 loads in flight against the WMMA chain.
// ---------------------------------------------------------------------------
__global__ __launch_bounds__(256) void k_feat_wmma(const float* __restrict__ h,
                                                   const float* __restrict__ Wt,
                                                   float* __restrict__ feat,
                                                   int N) {
  const int tid   = threadIdx.x;
  const int wave  = tid >> 5;
  const int lane  = tid & 31;
  const int mtile = wave >> 2;
  const int ntile = wave & 3;
  const int rowBase = blockIdx.x * 32 + mtile * 16;
  const int col   = ntile * 16 + (lane & 15);
  const int row   = rowBase + (lane & 15);
  const int khalf = (lane >> 4) << 1;   // 0 for lanes 0-15, 2 for lanes 16-31

  const int rowc = (row < N) ? row : (N - 1);
  const float* __restrict__ ap = h  + (size_t)rowc * IN_DIM + khalf;
  const float* __restrict__ bp = Wt + (size_t)col  * IN_DIM + khalf;

  v8f acc = {0.f, 0.f, 0.f, 0.f, 0.f, 0.f, 0.f, 0.f};

#pragma unroll
  for (int k = 0; k < IN_DIM; k += 16) {
    v2f a0, a1, a2, a3, b0, b1, b2, b3;
    a0 = *(const v2f*)(ap + k);      b0 = *(const v2f*)(bp + k);
    a1 = *(const v2f*)(ap + k + 4);  b1 = *(const v2f*)(bp + k + 4);
    a2 = *(const v2f*)(ap + k + 8);  b2 = *(const v2f*)(bp + k + 8);
    a3 = *(const v2f*)(ap + k + 12); b3 = *(const v2f*)(bp + k + 12);
    acc = __builtin_amdgcn_wmma_f32_16x16x4_f32(false, a0, false, b0, (short)0, acc, false, false);
    acc = __builtin_amdgcn_wmma_f32_16x16x4_f32(false, a1, false, b1, (short)0, acc, false, false);
    acc = __builtin_amdgcn_wmma_f32_16x16x4_f32(false, a2, false, b2, (short)0, acc, false, false);
    acc = __builtin_amdgcn_wmma_f32_16x16x4_f32(false, a3, false, b3, (short)0, acc, false, false);
  }

  const int mBase = rowBase + (lane >> 4) * 8;
#pragma unroll
  for (int r = 0; r < 8; ++r) {
    int m = mBase + r;
    if (m < N) feat[(size_t)m * OUT_DIM + col] = acc[r];
  }
}

// ---------------------------------------------------------------------------
// Kernel 2: per-node attention logits el = feat.attn_l, er = feat.attn_r.
// Also zeroes the per-destination edge-count array for the histogram.
// ---------------------------------------------------------------------------
__global__ void k_el_er(const float* __restrict__ feat,
                        const float* __restrict__ attn_l,
                        const float* __restrict__ attn_r,
                        float* __restrict__ el, float* __restrict__ er,
                        int* __restrict__ counts, int N) {
  int i = blockIdx.x * blockDim.x + threadIdx.x;
  if (i >= N) return;
  const float4* f  = (const float4*)(feat + (size_t)i * OUT_DIM);
  const float4* al = (const float4*)attn_l;
  const float4* ar = (const float4*)attn_r;
  float sl = 0.f, sr = 0.f;
#pragma unroll
  for (int j = 0; j < OUT_DIM / 4; ++j) {
    float4 v = f[j], a = al[j], b = ar[j];
    sl += v.x * a.x + v.y * a.y + v.z * a.z + v.w * a.w;
    sr += v.x * b.x + v.y * b.y + v.z * b.z + v.w * b.w;
  }
  el[i] = sl;
  er[i] = sr;
  counts[i] = 0;
}

// ---------------------------------------------------------------------------
// CSR build: histogram -> exclusive scan (2-level) -> bucket fill.
// ---------------------------------------------------------------------------
__global__ void k_hist(const int* __restrict__ dst, int* __restrict__ counts, int E) {
  int e = blockIdx.x * blockDim.x + threadIdx.x;
  if (e >= E) return;
  atomicAdd(&counts[dst[e]], 1);
}

__global__ __launch_bounds__(256) void k_scan1(const int* __restrict__ counts,
                                               int* __restrict__ offsets,
                                               int* __restrict__ bsum, int N) {
  __shared__ int sd[256];
  int tid  = threadIdx.x;
  int base = blockIdx.x * 1024 + tid * 4;
  int v0 = (base + 0 < N) ? counts[base + 0] : 0;
  int v1 = (base + 1 < N) ? counts[base + 1] : 0;
  int v2 = (base + 2 < N) ? counts[base + 2] : 0;
  int v3 = (base + 3 < N) ? counts[base + 3] : 0;
  int s = v0 + v1 + v2 + v3;
  sd[tid] = s;
  __syncthreads();
  for (int off = 1; off < 256; off <<= 1) {
    int t = (tid >= off) ? sd[tid - off] : 0;
    __syncthreads();
    sd[tid] += t;
    __syncthreads();
  }
  int excl = sd[tid] - s;
  if (base + 0 < N) offsets[base + 0] = excl;
  if (base + 1 < N) offsets[base + 1] = excl + v0;
  if (base + 2 < N) offsets[base + 2] = excl + v0 + v1;
  if (base + 3 < N) offsets[base + 3] = excl + v0 + v1 + v2;
  if (tid == 255) bsum[blockIdx.x] = sd[255];
}

__global__ void k_scan2(int* __restrict__ bsum, int nb) {
  if (threadIdx.x != 0 || blockIdx.x != 0) return;
  int running = 0;
  for (int b = 0; b < nb; ++b) {
    int t = bsum[b];
    bsum[b] = running;
    running += t;
  }
}

__global__ void k_scan3(int* __restrict__ offsets, const int* __restrict__ bsum,
                        int* __restrict__ counts, int N) {
  int i = blockIdx.x * blockDim.x + threadIdx.x;
  if (i >= N) return;
  offsets[i] += bsum[i >> 10];
  counts[i] = 0;  // becomes the per-destination cursor for k_fill
}

__global__ void k_fill(const int* __restrict__ src, const int* __restrict__ dst,
                       const int* __restrict__ offsets, int* __restrict__ cursor,
                       int* __restrict__ csr_src, int E) {
  int e = blockIdx.x * blockDim.x + threadIdx.x;
  if (e >= E) return;
  int d = dst[e];
  int p = offsets[d] + atomicAdd(&cursor[d], 1);
  csr_src[p] = src[e];
}

// ---------------------------------------------------------------------------
// Kernel 7: per-destination softmax + weighted aggregation. One wave32 per
// destination node; zero atomics, all gathers (feat is L2-resident).
// Lane L owns output channels 2L and 2L+1 -> single b64 gather per edge.
// Each lane's first-edge logit e0 is computed once and reused in all three
// passes (max / denom / alpha-broadcast via __shfl), since avg degree 16 < 32.
// ---------------------------------------------------------------------------
__global__ __launch_bounds__(256) void k_gat_agg(const float* __restrict__ feat,
                                                 const float* __restrict__ el,
                                                 const float* __restrict__ er,
                                                 const int* __restrict__ offsets,
                                                 const int* __restrict__ csr_src,
                                                 const float* __restrict__ bias,
                                                 float* __restrict__ out,
                                                 int N, int E) {
  int node = blockIdx.x * 8 + (threadIdx.x >> 5);
  if (node >= N) return;
  int lane  = threadIdx.x & 31;
  int start = offsets[node];
  int end   = (node == N - 1) ? E : offsets[node + 1];
  float er_d = er[node];

  // Lane's first edge logit (covers all edges when degree <= 32).
  int   j0 = start + lane;
  float e0 = (j0 < end) ? leaky(el[csr_src[j0]] + er_d) : -INFINITY;

  // Pass 1: segment max
  float m = e0;
  for (int j = j0 + 32; j < end; j += 32) {
    m = fmaxf(m, leaky(el[csr_src[j]] + er_d));
  }
#pragma unroll
  for (int off = 16; off > 0; off >>= 1) m = fmaxf(m, __shfl_xor(m, off, 32));
  if (end <= start) m = 0.f;  // zero-in-degree guard (matches isfinite guard)

  // Pass 2: denominator (reuse e0 for the first strip)
  float den = (j0 < end) ? __expf(e0 - m) : 0.f;
  for (int j = j0 + 32; j < end; j += 32) {
    den += __expf(leaky(el[csr_src[j]] + er_d) - m);
  }
#pragma unroll
  for (int off = 16; off > 0; off >>= 1) den += __shfl_xor(den, off, 32);
  float invden = (den > 0.f) ? 1.f / den : 1.f;

  // Pass 3: weighted sum; serial over edges, lanes over channel pairs.
  // Edge logits for the first 32 edges come from e0 via wave broadcast.
  float acc0 = 0.f, acc1 = 0.f;
  for (int j = start; j < end; ++j) {
    int t = j - start;                    // uniform across wave
    int s = csr_src[j];                   // uniform -> scalarized load
    float e;
    if (t < 32) {
      e = __shfl(e0, t, 32);              // broadcast cached logit
    } else {
      e = leaky(el[s] + er_d);
    }
    float alpha = __expf(e - m) * invden;
    v2f fv = *(const v2f*)(feat + (size_t)s * OUT_DIM + 2 * lane);
    acc0 += alpha * fv.x;
    acc1 += alpha * fv.y;
  }
  v2f bv = *(const v2f*)(bias + 2 * lane);
  v2f ov;
  ov.x = acc0 + bv.x;
  ov.y = acc1 + bv.y;
  *(v2f*)(out + (size_t)node * OUT_DIM + 2 * lane) = ov;
}

// ---------------------------------------------------------------------------
extern "C" void kernel_launch(void* const* d_in, const int* in_sizes, int n_in,
                              void* d_out, int out_size, void* d_ws, size_t ws_size,
                              hipStream_t stream) {
  const float* h      = (const float*)d_in[0];
  const int*   src    = (const int*)d_in[1];
  const int*   dst    = (const int*)d_in[2];
  const float* W      = (const float*)d_in[3];
  const float* attn_l = (const float*)d_in[4];
  const float* attn_r = (const float*)d_in[5];
  const float* bias   = (const float*)d_in[6];
  float* out = (float*)d_out;

  const int N = in_sizes[0] / IN_DIM;
  const int E = in_sizes[1];

  // Carve workspace (256B-aligned slices)
  char* ws = (char*)d_ws;
  size_t off = 0;
  auto carve = [&](size_t bytes) -> char* {
    char* p = ws + off;
    off = (off + bytes + 255) & ~(size_t)255;
    return p;
  };
  float* feat    = (float*)carve((size_t)N * OUT_DIM * sizeof(float));
  float* Wt      = (float*)carve((size_t)IN_DIM * OUT_DIM * sizeof(float));
  float* el      = (float*)carve((size_t)N * sizeof(float));
  float* er      = (float*)carve((size_t)N * sizeof(float));
  int*   counts  = (int*)  carve((size_t)N * sizeof(int));   // then cursor
  int*   offsets = (int*)  carve((size_t)N * sizeof(int));
  int*   bsum    = (int*)  carve(4096);
  int*   csr_src = (int*)  carve((size_t)E * sizeof(int));

  const int nb = (N + 1023) / 1024;

  // 0) Transpose W for contiguous B-fragment loads
  k_wt<<<(IN_DIM * OUT_DIM + 255) / 256, 256, 0, stream>>>(W, Wt);
  // 1) Projection (WMMA fp32)
  k_feat_wmma<<<(N + 31) / 32, 256, 0, stream>>>(h, Wt, feat, N);
  // 2) Node logits + zero counts
  k_el_er<<<(N + 255) / 256, 256, 0, stream>>>(feat, attn_l, attn_r, el, er, counts, N);
  // 3) CSR build
  k_hist <<<(E + 255) / 256, 256, 0, stream>>>(dst, counts, E);
  k_scan1<<<nb, 256, 0, stream>>>(counts, offsets, bsum, N);
  k_scan2<<<1, 1, 0, stream>>>(bsum, nb);
  k_scan3<<<(N + 255) / 256, 256, 0, stream>>>(offsets, bsum, counts, N);
  k_fill <<<(E + 255) / 256, 256, 0, stream>>>(src, dst, offsets, counts, csr_src, E);
  // 4) Softmax + aggregation (gather-based, atomic-free)
  k_gat_agg<<<(N + 7) / 8, 256, 0, stream>>>(feat, el, er, offsets, csr_src, bias, out, N, E);
}